// Controller_60816736911762
// MI455X (gfx1250) — compile-verified
//
#include <hip/hip_runtime.h>
#include <math.h>

typedef float v2f __attribute__((ext_vector_type(2)));
typedef float v4f __attribute__((ext_vector_type(4)));
typedef float v8f __attribute__((ext_vector_type(8)));

#define B_   256
#define T1_  2048
#define M_   128
#define D_   512
#define H_   1024
#define ROW_ (B_ * M_)      // 32768 floats per Val row
#define SEGS_   32
#define CHUNKS_ 16

__device__ __forceinline__ float sigm(float x) { return 1.0f / (1.0f + __expf(-x)); }

#define WMMA_F32(a, b, c) \
    __builtin_amdgcn_wmma_f32_16x16x4_f32(false, (a), false, (b), (short)0, (c), false, false)

// ---------------------------------------------------------------------------
// One GEMM operand stream: acc[0..3] (+)= A[16 x K] @ W^T tile.
// Software-pipelined with prefetch distance 2 (8 K-values resident, loads for
// the next 8 issued before the WMMAs execute) so the scheduler can use partial
// s_wait_loadcnt instead of a full drain before every v_wmma.
// Requires K % 8 == 0 and K >= 8.
// Fragment layouts per CDNA5 ISA 7.12.2:
//   A 16x4 : lane r<16 holds A[m0+r, k..k+1]; lane 16+r holds A[m0+r, k+2..k+3]
//   B 4x16 : lane r<16 holds B[k..k+1, n0+r] = W[n0+r, k..k+1]; lanes 16+ hold k+2..k+3
// ---------------------------------------------------------------------------
__device__ __forceinline__ void gemm_stream(
    v8f acc[4], const float* __restrict__ A, int lda, int K,
    const float* __restrict__ W, int ldw,
    int m0, int n0, int r, int half)
{
    const float* ap = A + (size_t)(m0 + r) * lda + 2 * half;
    const float* wp = W + (size_t)(n0 + r) * ldw + 2 * half;
    const size_t wstep = (size_t)16 * ldw;

    v2f aA = *(const v2f*)(ap + 0);
    v2f aB = *(const v2f*)(ap + 4);
    v2f bA[4], bB[4];
#pragma unroll
    for (int t = 0; t < 4; ++t) {
        bA[t] = *(const v2f*)(wp + wstep * t + 0);
        bB[t] = *(const v2f*)(wp + wstep * t + 4);
    }

    for (int k = 8; k < K; k += 8) {
        // issue next 10 loads before consuming the resident fragments
        v2f aC = *(const v2f*)(ap + k);
        v2f aD = *(const v2f*)(ap + k + 4);
        v2f bC[4], bD[4];
#pragma unroll
        for (int t = 0; t < 4; ++t) {
            bC[t] = *(const v2f*)(wp + wstep * t + k);
            bD[t] = *(const v2f*)(wp + wstep * t + k + 4);
        }
#pragma unroll
        for (int t = 0; t < 4; ++t) acc[t] = WMMA_F32(aA, bA[t], acc[t]);
#pragma unroll
        for (int t = 0; t < 4; ++t) acc[t] = WMMA_F32(aB, bB[t], acc[t]);
        aA = aC;
        aB = aD;
#pragma unroll
        for (int t = 0; t < 4; ++t) { bA[t] = bC[t]; bB[t] = bD[t]; }
    }
    // drain
#pragma unroll
    for (int t = 0; t < 4; ++t) acc[t] = WMMA_F32(aA, bA[t], acc[t]);
#pragma unroll
    for (int t = 0; t < 4; ++t) acc[t] = WMMA_F32(aB, bB[t], acc[t]);
}

// ---------------------------------------------------------------------------
// C[Mrows x N] = A1 @ W1^T + A2 @ W2^T + bias1 + bias2
// W row-major [N, ldw]; C[b,n] = sum_k A[b,k]*W[n,k].
// One wave computes a 16x64 tile of C.
//   C/D layout: VGPR v -> row m0+v (lanes 0-15) / m0+v+8 (lanes 16-31), lane%16 = col
// ---------------------------------------------------------------------------
__global__ __launch_bounds__(32) void wmma_dual_gemm(
    const float* __restrict__ A1, int lda1, int K1,
    const float* __restrict__ W1, int ldw1,
    const float* __restrict__ A2, int lda2, int K2,
    const float* __restrict__ W2, int ldw2,
    const float* __restrict__ bias1, const float* __restrict__ bias2,
    float* __restrict__ C, int ldc)
{
    const int lane = threadIdx.x;
    const int r    = lane & 15;
    const int half = lane >> 4;
    const int m0   = blockIdx.y << 4;
    const int n0   = blockIdx.x << 6;

    v8f acc[4];
#pragma unroll
    for (int t = 0; t < 4; ++t) {
        const int col = n0 + 16 * t + r;
        float bvv = 0.0f;
        if (bias1) bvv += bias1[col];
        if (bias2) bvv += bias2[col];
#pragma unroll
        for (int e = 0; e < 8; ++e) acc[t][e] = bvv;   // whole VGPR shares one column
    }

    gemm_stream(acc, A1, lda1, K1, W1, ldw1, m0, n0, r, half);
    if (A2) gemm_stream(acc, A2, lda2, K2, W2, ldw2, m0, n0, r, half);

#pragma unroll
    for (int t = 0; t < 4; ++t) {
        const int col = n0 + 16 * t + r;
#pragma unroll
        for (int v = 0; v < 8; ++v) {
            C[(size_t)(m0 + v + 8 * half) * ldc + col] = acc[t][v];
        }
    }
}

// ---------------------------------------------------------------------------
// LSTM gate nonlinearity: g [B,4H] (i|f|g|o), c_prev [B,H] -> h, c
// ---------------------------------------------------------------------------
__global__ __launch_bounds__(256) void lstm_gates(
    const float* __restrict__ g, const float* __restrict__ c_prev,
    float* __restrict__ h_ws, float* __restrict__ h_out, float* __restrict__ c_out)
{
    const int e = blockIdx.x * blockDim.x + threadIdx.x;   // < B*H
    const int b = e >> 10;
    const int j = e & (H_ - 1);
    const float* gr = g + (size_t)b * (4 * H_);
    const float gi = gr[j];
    const float gf = gr[j + H_];
    const float gg = gr[j + 2 * H_];
    const float go = gr[j + 3 * H_];
    const float c  = sigm(gf) * c_prev[e] + sigm(gi) * tanhf(gg);
    const float hh = sigm(go) * tanhf(c);
    h_ws[e]  = hh;
    h_out[e] = hh;
    c_out[e] = c;
}

__global__ __launch_bounds__(256) void tanh_ew(
    const float* __restrict__ src, float* __restrict__ dst, int n)
{
    const int e = blockIdx.x * blockDim.x + threadIdx.x;
    if (e < n) dst[e] = tanhf(src[e]);
}

// ---------------------------------------------------------------------------
// dt/ut heads: one wave per batch row, two 1024-dot-products, wave reduction
// ---------------------------------------------------------------------------
__global__ __launch_bounds__(256) void head_du(
    const float* __restrict__ h,
    const float* __restrict__ Wd, const float* __restrict__ bd,
    const float* __restrict__ Wu, const float* __restrict__ bu,
    float* __restrict__ dt, float* __restrict__ ut)
{
    const int wave = (blockIdx.x * blockDim.x + threadIdx.x) >> 5;   // == b
    const int lane = threadIdx.x & 31;
    const float* hp = h + (size_t)wave * H_;
    float sd = 0.0f, su = 0.0f;
    for (int j = lane; j < H_; j += 32) {
        const float hv = hp[j];
        sd += hv * Wd[j];
        su += hv * Wu[j];
    }
#pragma unroll
    for (int off = 16; off > 0; off >>= 1) {
        sd += __shfl_xor(sd, off, 32);
        su += __shfl_xor(su, off, 32);
    }
    if (lane == 0) {
        dt[wave] = sigm(sd + bd[0]);
        ut[wave] = sigm(su + bu[0]);
    }
}

// ---------------------------------------------------------------------------
// Fused push-scan + read-coefficient scan. Both walk i downward, so one pass:
//   stg[i] = min(prev_stg[i] - min(u,0), 0); u -= stg[i]           (push)
//   coef[i] = min(stg[i], min(read,0));      read -= stg[i]        (read)
// starting at i = T1 with stg[T1] = dt. One lane per batch element.
// ---------------------------------------------------------------------------
__global__ __launch_bounds__(256) void stack_scan(
    const float* __restrict__ prev_stg,
    const float* __restrict__ dt, const float* __restrict__ ut,
    float* __restrict__ stg_out, float* __restrict__ coef)
{
    const int b = threadIdx.x;
    float u    = ut[b];
    float read = 1.0f;
    {
        const float s = dt[b];
        stg_out[(size_t)T1_ * B_ + b] = s;
        coef[(size_t)T1_ * B_ + b]    = fminf(s, fminf(read, 0.0f));
        read -= s;
    }
    for (int i = T1_ - 1; i >= 0; --i) {
        const float ps = prev_stg[(size_t)i * B_ + b];
        const float s  = fminf(ps - fminf(u, 0.0f), 0.0f);
        u -= s;
        stg_out[(size_t)i * B_ + b] = s;
        const float cf = fminf(s, fminf(read, 0.0f));
        coef[(size_t)i * B_ + b] = cf;
        read -= s;
    }
}

// ---------------------------------------------------------------------------
// Fused Val copy + rt accumulation. Streams 268MB once with non-temporal
// loads/stores (keep L2 for weights/coef), accumulating coef[i,b]*Val[i,b,m]
// into per-chunk partials (deterministic; no float atomics).
// ---------------------------------------------------------------------------
__global__ __launch_bounds__(256) void val_copy_acc(
    const float* __restrict__ prevVal, const float* __restrict__ coef,
    float* __restrict__ outVal, float* __restrict__ partial)
{
    const int seg   = blockIdx.x;             // 0..31, 1024-float column segment
    const int chunk = blockIdx.y;             // 0..15, 128-row i chunk
    const int col   = seg * 1024 + threadIdx.x * 4;
    const int b     = col >> 7;               // M_ = 128
    const int i0    = chunk * (T1_ / CHUNKS_);

    v4f acc = {0.0f, 0.0f, 0.0f, 0.0f};
    const float* src = prevVal + (size_t)i0 * ROW_ + col;
    float*       dst = outVal  + (size_t)i0 * ROW_ + col;
    const float* cp  = coef    + (size_t)i0 * B_   + b;

#pragma unroll 4
    for (int it = 0; it < T1_ / CHUNKS_; ++it) {
        __builtin_prefetch(src + (size_t)8 * ROW_, 0, 1);   // global_prefetch_b8
        v4f v = __builtin_nontemporal_load((const v4f*)src);
        __builtin_nontemporal_store(v, (v4f*)dst);
        acc += (*cp) * v;
        src += ROW_;
        dst += ROW_;
        cp  += B_;
    }
    if (chunk == CHUNKS_ - 1) {   // row T1 = vt, already written by tanh_ew
        v4f v = *(const v4f*)(outVal + (size_t)T1_ * ROW_ + col);
        acc += coef[(size_t)T1_ * B_ + b] * v;
    }
    *(v4f*)(partial + (size_t)chunk * ROW_ + col) = acc;
}

__global__ __launch_bounds__(256) void reduce_partials(
    const float* __restrict__ partial, float* __restrict__ rt)
{
    const int e = blockIdx.x * blockDim.x + threadIdx.x;   // < ROW_
    float s = 0.0f;
#pragma unroll
    for (int c = 0; c < CHUNKS_; ++c) s += partial[(size_t)c * ROW_ + e];
    rt[e] = s;
}

// ---------------------------------------------------------------------------
extern "C" void kernel_launch(void* const* d_in, const int* in_sizes, int n_in,
                              void* d_out, int out_size, void* d_ws, size_t ws_size,
                              hipStream_t stream)
{
    const float* x         = (const float*)d_in[0];
    const float* prev_Val  = (const float*)d_in[1];
    const float* prev_stg  = (const float*)d_in[2];
    const float* h0        = (const float*)d_in[3];
    const float* c0        = (const float*)d_in[4];
    const float* prev_read = (const float*)d_in[5];
    const float* Wproj     = (const float*)d_in[6];
    const float* Wih0      = (const float*)d_in[7];
    const float* Whh0      = (const float*)d_in[8];
    const float* bih0      = (const float*)d_in[9];
    const float* bhh0      = (const float*)d_in[10];
    const float* Wih1      = (const float*)d_in[11];
    const float* Whh1      = (const float*)d_in[12];
    const float* bih1      = (const float*)d_in[13];
    const float* bhh1      = (const float*)d_in[14];
    const float* Wd        = (const float*)d_in[15];
    const float* bd        = (const float*)d_in[16];
    const float* Wu        = (const float*)d_in[17];
    const float* bu        = (const float*)d_in[18];
    const float* Wv        = (const float*)d_in[19];
    const float* bv        = (const float*)d_in[20];
    const float* Wo        = (const float*)d_in[21];
    const float* bo        = (const float*)d_in[22];

    // output layout (flat, return order): ot, Val, stg, hidden_h, hidden_c, rt
    float* out      = (float*)d_out;
    float* out_ot   = out;                                     // 256*512
    float* out_Val  = out_ot + (size_t)B_ * D_;                // 2049*256*128
    float* out_stg  = out_Val + (size_t)(T1_ + 1) * ROW_;      // 2049*256
    float* out_hh   = out_stg + (size_t)(T1_ + 1) * B_;        // 2*256*1024
    float* out_hc   = out_hh + (size_t)2 * B_ * H_;            // 2*256*1024
    float* out_rt   = out_hc + (size_t)2 * B_ * H_;            // 256*128

    // workspace layout (floats)
    float* ws       = (float*)d_ws;
    float* ws_xaug  = ws;                                  // 131072
    float* ws_g     = ws_xaug + (size_t)B_ * D_;           // 1048576
    float* ws_h0n   = ws_g + (size_t)B_ * 4 * H_;          // 262144
    float* ws_h1n   = ws_h0n + (size_t)B_ * H_;            // 262144
    float* ws_vtp   = ws_h1n + (size_t)B_ * H_;            // 32768
    float* ws_otp   = ws_vtp + (size_t)B_ * M_;            // 131072
    float* ws_dt    = ws_otp + (size_t)B_ * D_;            // 256
    float* ws_ut    = ws_dt + B_;                          // 256
    float* ws_coef  = ws_ut + B_;                          // 2049*256
    float* ws_part  = ws_coef + (size_t)(T1_ + 1) * B_;    // 16*32768

    const dim3 wave(32, 1, 1);

    // 1. x_aug = x @ Wproj[:, :512]^T + prev_read @ Wproj[:, 512:]^T
    wmma_dual_gemm<<<dim3(D_ / 64, B_ / 16), wave, 0, stream>>>(
        x, D_, D_, Wproj, D_ + M_,
        prev_read, M_, M_, Wproj + D_, D_ + M_,
        nullptr, nullptr, ws_xaug, D_);

    // 2. LSTM layer 0: g = x_aug @ Wih0^T + h0[0] @ Whh0^T + bih0 + bhh0
    wmma_dual_gemm<<<dim3(4 * H_ / 64, B_ / 16), wave, 0, stream>>>(
        ws_xaug, D_, D_, Wih0, D_,
        h0, H_, H_, Whh0, H_,
        bih0, bhh0, ws_g, 4 * H_);
    lstm_gates<<<dim3(B_ * H_ / 256), dim3(256), 0, stream>>>(
        ws_g, c0, ws_h0n, out_hh, out_hc);

    // 3. LSTM layer 1
    wmma_dual_gemm<<<dim3(4 * H_ / 64, B_ / 16), wave, 0, stream>>>(
        ws_h0n, H_, H_, Wih1, H_,
        h0 + (size_t)B_ * H_, H_, H_, Whh1, H_,
        bih1, bhh1, ws_g, 4 * H_);
    lstm_gates<<<dim3(B_ * H_ / 256), dim3(256), 0, stream>>>(
        ws_g, c0 + (size_t)B_ * H_, ws_h1n,
        out_hh + (size_t)B_ * H_, out_hc + (size_t)B_ * H_);

    // 4. heads: vt = tanh(h1n@Wv^T+bv) -> Val[T1]; ot = tanh(h1n@Wo^T+bo)
    wmma_dual_gemm<<<dim3(M_ / 64, B_ / 16), wave, 0, stream>>>(
        ws_h1n, H_, H_, Wv, H_,
        nullptr, 0, 0, nullptr, 0, bv, nullptr, ws_vtp, M_);
    wmma_dual_gemm<<<dim3(D_ / 64, B_ / 16), wave, 0, stream>>>(
        ws_h1n, H_, H_, Wo, H_,
        nullptr, 0, 0, nullptr, 0, bo, nullptr, ws_otp, D_);
    tanh_ew<<<dim3(B_ * M_ / 256), dim3(256), 0, stream>>>(
        ws_vtp, out_Val + (size_t)T1_ * ROW_, B_ * M_);
    tanh_ew<<<dim3(B_ * D_ / 256), dim3(256), 0, stream>>>(ws_otp, out_ot, B_ * D_);
    head_du<<<dim3(B_ / 8), dim3(256), 0, stream>>>(
        ws_h1n, Wd, bd, Wu, bu, ws_dt, ws_ut);

    // 5. fused scans -> stg (output) + coef (workspace)
    stack_scan<<<dim3(1), dim3(256), 0, stream>>>(
        prev_stg, ws_dt, ws_ut, out_stg, ws_coef);

    // 6. fused Val copy + rt partial accumulation, then deterministic reduce
    val_copy_acc<<<dim3(SEGS_, CHUNKS_), dim3(256), 0, stream>>>(
        prev_Val, ws_coef, out_Val, ws_part);
    reduce_partials<<<dim3(ROW_ / 256), dim3(256), 0, stream>>>(ws_part, out_rt);
}